// MessagePassingLayer_29506425324200
// MI455X (gfx1250) — compile-verified
//
#include <hip/hip_runtime.h>
#include <hip/hip_bf16.h>

#define N_NODES 50000
#define N_EDGES 800000
#define DIM 128
#define BN_EPS 1e-5f

typedef __attribute__((ext_vector_type(2))) float v2f;
typedef __attribute__((ext_vector_type(8))) float v8f;

__device__ __forceinline__ float sigmoidf_(float x) { return 1.0f / (1.0f + __expf(-x)); }

__device__ __forceinline__ v8f wmma4(v2f a, v2f b, v8f c) {
    // D = A(16x4,f32) * B(4x16,f32) + C(16x16,f32), exact fp32
    return __builtin_amdgcn_wmma_f32_16x16x4_f32(false, a, false, b, (short)0, c, false, false);
}

// ---------------- zero workspace (agg + a_src + a_dst + stats) ----------------
__global__ void kZero(float* __restrict__ p, long n) {
    long i = (long)blockIdx.x * blockDim.x + threadIdx.x;
    long s = (long)gridDim.x * blockDim.x;
    for (; i < n; i += s) p[i] = 0.0f;
}

// ---------------- per-node attention dot products ----------------
// a_src[n] = x[n] . W_att[0:128],  a_dst[n] = x[n] . W_att[128:256]
__global__ __launch_bounds__(256) void kAtt(const float* __restrict__ x,
                                            const float* __restrict__ Watt,
                                            float* __restrict__ a_src,
                                            float* __restrict__ a_dst) {
    int lane = threadIdx.x & 31;
    int wave = threadIdx.x >> 5;
    int node = blockIdx.x * 8 + wave;           // grid = N/8 exactly
    const float* xr = x + (long)node * DIM;
    float s = 0.0f, d = 0.0f;
#pragma unroll
    for (int j = 0; j < 4; ++j) {
        float xv = xr[lane + 32 * j];
        s += xv * Watt[lane + 32 * j];
        d += xv * Watt[DIM + lane + 32 * j];
    }
#pragma unroll
    for (int off = 16; off > 0; off >>= 1) {
        s += __shfl_down(s, off);
        d += __shfl_down(d, off);
    }
    if (lane == 0) { a_src[node] = s; a_dst[node] = d; }
}

// ---------------- per-node message GEMM: m = x @ W_msg + b_msg ----------------
// block = 256 threads = 8 waves; wave w -> N-tile w (16 cols), blockIdx -> M-tile (16 nodes)
__global__ __launch_bounds__(256) void kMsg(const float* __restrict__ x,
                                            const float* __restrict__ Wmsg,
                                            const float* __restrict__ bmsg,
                                            float* __restrict__ m) {
    const int lane = threadIdx.x & 31;
    const int wave = threadIdx.x >> 5;
    const int m0 = blockIdx.x * 16;
    const int n0 = wave * 16;
    const int row = lane & 15;
    const int hi  = lane >> 4;        // 0: K={k,k+1}, 1: K={k+2,k+3}
    const int koff = hi * 2;
    const int col = lane & 15;

    v8f c = {};
    const float* xrow = x + (long)(m0 + row) * DIM + koff;
    const float* wcol = Wmsg + n0 + col;
#pragma unroll 4
    for (int k = 0; k < DIM; k += 4) {
        v2f a = *(const v2f*)(xrow + k);
        v2f b;
        b.x = wcol[(k + koff) * DIM];
        b.y = wcol[(k + koff + 1) * DIM];
        c = wmma4(a, b, c);
    }
    float bias = bmsg[n0 + col];
#pragma unroll
    for (int r = 0; r < 8; ++r) {
        int gr = m0 + r + 8 * hi;
        m[(long)gr * DIM + n0 + col] = c[r] + bias;
    }
}

// ---------------- edge scatter: agg[dst] += sigmoid(as+ad+b) * m[src] ----------------
// one wave per edge; lane loads float4 of the message, 4 hardware f32 atomics
__global__ __launch_bounds__(256) void kEdge(const int* __restrict__ ei,
                                             const float* __restrict__ a_src,
                                             const float* __restrict__ a_dst,
                                             const float* __restrict__ batt,
                                             const float* __restrict__ m,
                                             float* __restrict__ agg) {
    int e = blockIdx.x * 8 + (threadIdx.x >> 5);    // grid = E/8 exactly
    int lane = threadIdx.x & 31;
    int src = ei[e];
    int dst = ei[N_EDGES + e];
    float att = sigmoidf_(a_src[src] + a_dst[dst] + batt[0]);
    const float4 v = *(const float4*)(m + (long)src * DIM + lane * 4);
    float* o = agg + (long)dst * DIM + lane * 4;
    // force native global_atomic_add_f32 (no CAS loop) — this is the hot path
    unsafeAtomicAdd(o + 0, v.x * att);
    unsafeAtomicAdd(o + 1, v.y * att);
    unsafeAtomicAdd(o + 2, v.z * att);
    unsafeAtomicAdd(o + 3, v.w * att);
}

// ---------------- fused GRU: gi = agg@W_ih^T+b_ih, gh = x@W_hh^T+b_hh, gates ----------------
// wave -> 16-feature tile f0, blockIdx -> 16-node tile; 6 WMMA accumulators
__global__ __launch_bounds__(256) void kGru(const float* __restrict__ agg,
                                            const float* __restrict__ x,
                                            const float* __restrict__ Wih,
                                            const float* __restrict__ bih,
                                            const float* __restrict__ Whh,
                                            const float* __restrict__ bhh,
                                            float* __restrict__ hout) {
    const int lane = threadIdx.x & 31;
    const int wave = threadIdx.x >> 5;
    const int m0 = blockIdx.x * 16;
    const int f0 = wave * 16;
    const int row = lane & 15;
    const int hi  = lane >> 4;
    const int koff = hi * 2;
    const int col = lane & 15;

    v8f cir = {}, ciz = {}, cin = {}, chr_ = {}, chz = {}, chn = {};
    const float* aggRow = agg + (long)(m0 + row) * DIM + koff;
    const float* xRow   = x   + (long)(m0 + row) * DIM + koff;
    // B[k][j] = W[j][k]; W row-major [3*DIM, DIM] -> consecutive k is contiguous (float2 loads)
    const float* bi0 = Wih + (long)(0 * DIM + f0 + col) * DIM + koff;
    const float* bi1 = Wih + (long)(1 * DIM + f0 + col) * DIM + koff;
    const float* bi2 = Wih + (long)(2 * DIM + f0 + col) * DIM + koff;
    const float* bh0 = Whh + (long)(0 * DIM + f0 + col) * DIM + koff;
    const float* bh1 = Whh + (long)(1 * DIM + f0 + col) * DIM + koff;
    const float* bh2 = Whh + (long)(2 * DIM + f0 + col) * DIM + koff;

#pragma unroll 2
    for (int k = 0; k < DIM; k += 4) {
        v2f aA = *(const v2f*)(aggRow + k);
        v2f aX = *(const v2f*)(xRow + k);
        cir  = wmma4(aA, *(const v2f*)(bi0 + k), cir);
        ciz  = wmma4(aA, *(const v2f*)(bi1 + k), ciz);
        cin  = wmma4(aA, *(const v2f*)(bi2 + k), cin);
        chr_ = wmma4(aX, *(const v2f*)(bh0 + k), chr_);
        chz  = wmma4(aX, *(const v2f*)(bh1 + k), chz);
        chn  = wmma4(aX, *(const v2f*)(bh2 + k), chn);
    }
    float bir = bih[f0 + col], biz = bih[DIM + f0 + col], bin = bih[2 * DIM + f0 + col];
    float bhr = bhh[f0 + col], bhz = bhh[DIM + f0 + col], bhn = bhh[2 * DIM + f0 + col];
#pragma unroll
    for (int r = 0; r < 8; ++r) {
        int gr = m0 + r + 8 * hi;
        float xv = x[(long)gr * DIM + f0 + col];
        float rr = sigmoidf_((cir[r] + bir) + (chr_[r] + bhr));
        float zz = sigmoidf_((ciz[r] + biz) + (chz[r] + bhz));
        float nn = tanhf((cin[r] + bin) + rr * (chn[r] + bhn));
        hout[(long)gr * DIM + f0 + col] = (1.0f - zz) * nn + zz * xv;
    }
}

// ---------------- BN statistics: stats[f]=sum, stats[128+f]=sumsq ----------------
__global__ __launch_bounds__(256) void kBnSum(const float* __restrict__ h,
                                              float* __restrict__ stats) {
    __shared__ float ssum[256], ssq[256];
    int tid = threadIdx.x;
    int f = tid & 127, half = tid >> 7;
    float s = 0.0f, q = 0.0f;
    for (int row = blockIdx.x * 2 + half; row < N_NODES; row += gridDim.x * 2) {
        float v = h[(long)row * DIM + f];
        s += v; q += v * v;
    }
    ssum[tid] = s; ssq[tid] = q;
    __syncthreads();
    if (tid < 128) {
        atomicAdd(&stats[f],       ssum[tid] + ssum[tid + 128]);
        atomicAdd(&stats[128 + f], ssq[tid]  + ssq[tid + 128]);
    }
}

// ---------------- BN apply ----------------
__global__ __launch_bounds__(256) void kBnApply(const float* __restrict__ h,
                                                const float* __restrict__ stats,
                                                const float* __restrict__ gamma,
                                                const float* __restrict__ beta,
                                                float* __restrict__ out) {
    const float invN = 1.0f / (float)N_NODES;
    long total = (long)N_NODES * DIM;
    for (long i = (long)blockIdx.x * blockDim.x + threadIdx.x; i < total;
         i += (long)gridDim.x * blockDim.x) {
        int f = (int)(i & 127);
        float mean = stats[f] * invN;
        float var  = stats[128 + f] * invN - mean * mean;
        out[i] = gamma[f] * (h[i] - mean) * rsqrtf(var + BN_EPS) + beta[f];
    }
}

extern "C" void kernel_launch(void* const* d_in, const int* in_sizes, int n_in,
                              void* d_out, int out_size, void* d_ws, size_t ws_size,
                              hipStream_t stream) {
    const float* x    = (const float*)d_in[0];
    const int*   ei   = (const int*)d_in[1];   // edge_index (JAX default x64-off -> int32)
    const float* Wmsg = (const float*)d_in[2];
    const float* bmsg = (const float*)d_in[3];
    const float* Watt = (const float*)d_in[4];
    const float* batt = (const float*)d_in[5];
    const float* Wih  = (const float*)d_in[6];
    const float* bih  = (const float*)d_in[7];
    const float* Whh  = (const float*)d_in[8];
    const float* bhh  = (const float*)d_in[9];
    const float* gamma = (const float*)d_in[10];
    const float* beta  = (const float*)d_in[11];
    float* out = (float*)d_out;

    // workspace layout (floats): [mh: N*128][agg: N*128][a_src: N][a_dst: N][stats: 256]
    float* ws    = (float*)d_ws;
    float* mh    = ws;                                 // messages, later reused for h_new
    float* agg   = mh  + (size_t)N_NODES * DIM;
    float* asrc  = agg + (size_t)N_NODES * DIM;
    float* adst  = asrc + N_NODES;
    float* stats = adst + N_NODES;

    // zero agg..stats (a_src/a_dst harmlessly included; overwritten by kAtt)
    long zn = (long)N_NODES * DIM + 2L * N_NODES + 256;
    kZero<<<2048, 256, 0, stream>>>(agg, zn);

    kAtt<<<N_NODES / 8, 256, 0, stream>>>(x, Watt, asrc, adst);
    kMsg<<<N_NODES / 16, 256, 0, stream>>>(x, Wmsg, bmsg, mh);
    kEdge<<<N_EDGES / 8, 256, 0, stream>>>(ei, asrc, adst, batt, mh, agg);
    kGru<<<N_NODES / 16, 256, 0, stream>>>(agg, x, Wih, bih, Whh, bhh, mh);
    kBnSum<<<512, 256, 0, stream>>>(mh, stats);
    kBnApply<<<2048, 256, 0, stream>>>(mh, stats, gamma, beta, out);
}